// ImageUpgradingBlock_87522843560728
// MI455X (gfx1250) — compile-verified
//
#include <hip/hip_runtime.h>

typedef __attribute__((ext_vector_type(2))) float v2f;
typedef __attribute__((ext_vector_type(8))) float v8f;

#define BATCH  3072
#define HW     4096
#define KDIM   12288          // 64*64*3
#define NCHUNK (KDIM / 4)     // 3072 k-chunks of 4
#define OUTB   2048

// ws layout in floats
#define WS_BPACK 0                          // 3072*64   = 196608 floats
#define WS_VBUF  196608                     // 3072*6    = 18432
#define WS_MBUF  (196608 + 18432)           // 3072*6    = 18432
#define WS_K     (196608 + 18432 + 18432)   // 64*64     = 4096
// total < 1 MB of ws

// ---- pack W1|W2 into WMMA-B layout: chunk-major, per chunk 32 lanes x 2 comps
// B element (VGPR v, lane l): K = chunk*4 + v + (l>=16 ? 2 : 0), N = l & 15
__global__ void k_pack_w(const float* __restrict__ w1, const float* __restrict__ w2,
                         float* __restrict__ bp) {
    int t = blockIdx.x * blockDim.x + threadIdx.x;
    if (t >= NCHUNK * 64) return;
    int chunk = t >> 6;
    int r = t & 63;
    int lane = r >> 1;
    int v = r & 1;
    int k = chunk * 4 + v + (lane >= 16 ? 2 : 0);
    int N = lane & 15;
    float val = 0.0f;
    if (N < 3)      val = w1[k * 3 + N];
    else if (N < 6) val = w2[k * 3 + (N - 3)];
    bp[t] = val;
}

// seed vbuf with biases (k_proj atomically accumulates on top)
__global__ void k_init_v(const float* __restrict__ b1, const float* __restrict__ b2,
                         float* __restrict__ vbuf) {
    int t = blockIdx.x * blockDim.x + threadIdx.x;
    if (t >= BATCH * 6) return;
    int N = t % 6;
    vbuf[t] = (N < 3) ? b1[N] : b2[N - 3];
}

// Hilbert kernel matrix: K[i,j] = (1/32) * sum_{f=1..31} sin(pi*f*(i-j)/32)
__global__ void k_init_K(float* __restrict__ Kg) {
    int t = blockIdx.x * blockDim.x + threadIdx.x;
    if (t >= 4096) return;
    int i = t >> 6, j = t & 63;
    float m = (float)(i - j);
    float s = 0.0f;
    for (int f = 1; f <= 31; ++f)
        s += sinf(3.14159265358979323846f * (float)f * m / 32.0f);
    Kg[t] = s * (1.0f / 32.0f);
}

// ---- phase 1: v = X @ [W1|W2] via f32 WMMA 16x16x4 ----------------------
// grid (BATCH/16, 4): block = 16 batches, y = quarter of k; 8 waves split k.
__global__ void __launch_bounds__(256) k_proj(const float* __restrict__ X,
                                              const float* __restrict__ bp,
                                              float* __restrict__ vbuf) {
    int b0   = blockIdx.x * 16;
    int wave = threadIdx.x >> 5;
    int lane = threadIdx.x & 31;
    int kstart = blockIdx.y * 3072 + wave * 384;   // 96 chunks per wave
    int chunk0 = kstart >> 2;

    // A (16x4 f32): lane<16 -> row=lane, K=kb..kb+1 ; lane>=16 -> row=lane-16, K=kb+2..kb+3
    const float* arow = X + (size_t)(b0 + (lane & 15)) * KDIM + (lane >= 16 ? 2 : 0);
    const float* brow = bp + (size_t)chunk0 * 64 + lane * 2;

    v8f c = {};
    for (int kc = 0; kc < 96; ++kc) {
        v2f a = *(const v2f*)(arow + kstart + kc * 4);
        v2f b = *(const v2f*)(brow + kc * 64);
        c = __builtin_amdgcn_wmma_f32_16x16x4_f32(false, a, false, b, (short)0, c,
                                                  false, false);
    }

    __shared__ float red[8][32][8];   // 8 KB
    #pragma unroll
    for (int j = 0; j < 8; ++j) red[wave][lane][j] = c[j];
    __syncthreads();

    // cross-wave reduction: thread t handles one (lane, vgpr) cell
    int l = threadIdx.x & 31;
    int j = threadIdx.x >> 5;
    float s = 0.0f;
    #pragma unroll
    for (int w = 0; w < 8; ++w) s += red[w][l][j];
    int M = j + (l >= 16 ? 8 : 0);   // C/D layout: vgpr j holds rows j and j+8
    int N = l & 15;
    if (N < 6) atomicAdd(&vbuf[(b0 + M) * 6 + N], s);
}

// ---- phase 2: per-batch normalize, 2x2 inverse, M = Ginv * V^T ----------
__global__ void k_small(const float* __restrict__ vbuf, float* __restrict__ Mbuf) {
    int b = blockIdx.x * blockDim.x + threadIdx.x;
    if (b >= BATCH) return;
    float u[3], v[3];
    #pragma unroll
    for (int cc = 0; cc < 3; ++cc) {
        u[cc] = vbuf[b * 6 + cc];
        v[cc] = vbuf[b * 6 + 3 + cc];
    }
    float nu = 1e-6f + fabsf(u[0]) + fabsf(u[1]) + fabsf(u[2]);
    float nv = 1e-6f + fabsf(v[0]) + fabsf(v[1]) + fabsf(v[2]);
    #pragma unroll
    for (int cc = 0; cc < 3; ++cc) { u[cc] /= nu; v[cc] /= nv; }
    float a  = u[0]*u[0] + u[1]*u[1] + u[2]*u[2];
    float bb = u[0]*v[0] + u[1]*v[1] + u[2]*v[2];
    float d  = v[0]*v[0] + v[1]*v[1] + v[2]*v[2];
    float inv = 1.0f / (a * d - bb * bb);
    #pragma unroll
    for (int cc = 0; cc < 3; ++cc) {
        Mbuf[b * 6 + cc]     = ( d * u[cc] - bb * v[cc]) * inv;  // row k=0
        Mbuf[b * 6 + 3 + cc] = (-bb * u[cc] + a * v[cc]) * inv;  // row k=1
    }
}

// ---- fused phase 3+4: build rec tile for one output image in LDS,
//      write channels 0..2, then rec_h = K @ rec via WMMA -> channels 3..5.
// Output image b' covers rec flat elements e in [b'*12288, (b'+1)*12288),
// and e equals the flat z index f = b*8192 + n*2 + k  (b = input batch).
__global__ void __launch_bounds__(256) k_rec_hilbert(const float* __restrict__ X,
                                                     const float* __restrict__ Mbuf,
                                                     const float* __restrict__ Kg,
                                                     float* __restrict__ out) {
    __shared__ float recs[64 * 192];   // 48 KB: rec tile, flat t = h*192 + (w*3+c)
    int bimg = blockIdx.x;
    size_t base  = (size_t)bimg * 64 * 64 * 6;   // out image base
    long   ebase = (long)bimg * 12288;           // rec flat base

    // ---- phase A: compute rec pairs straight from X ----------------------
    // pair p handles rec elements t=2p, 2p+1  (one input pixel, both k rows)
    for (int p = threadIdx.x; p < 6144; p += 256) {
        int  t = p * 2;
        long f = ebase + t;                 // global flat z index
        int  b = (int)(f >> 13);            // input batch (8192 per batch)
        int  r = (int)(f & 8191);
        int  n = r >> 1;                    // input pixel
        const float* m = Mbuf + b * 6;
        const float* x = X + (size_t)b * KDIM + n * 3;
        float x0 = x[0], x1 = x[1], x2 = x[2];
        float z0 = m[0] * x0 + m[1] * x1 + m[2] * x2;
        float z1 = m[3] * x0 + m[4] * x1 + m[5] * x2;
        recs[t]     = z0;
        recs[t + 1] = z1;
        // rec also goes to out channels 0..2:  t -> pixel t/3, chan t%3
        out[base + (size_t)(t / 3) * 6 + (t % 3)]             = z0;
        out[base + (size_t)((t + 1) / 3) * 6 + ((t + 1) % 3)] = z1;
    }
    __syncthreads();

    // ---- phase B: rec_h = K (64x64) @ rec (64x192) via f32 WMMA ----------
    int wave = threadIdx.x >> 5;
    int lane = threadIdx.x & 31;
    int joff = (lane >= 16 ? 2 : 0);
    int ln15 = lane & 15;

    for (int tt = wave; tt < 48; tt += 8) {   // 4 x 12 tiles of 16x16
        int i0 = (tt / 12) * 16;
        int n0 = (tt % 12) * 16;
        int ia  = i0 + ln15;
        int col = n0 + ln15;
        v8f c = {};
        for (int jb = 0; jb < 64; jb += 4) {
            v2f a = *(const v2f*)(Kg + ia * 64 + jb + joff);   // A: Hilbert matrix tile
            int jx = jb + joff;
            v2f bm;
            bm.x = recs[jx * 192 + col];                       // B: rec rows jx, jx+1
            bm.y = recs[(jx + 1) * 192 + col];
            c = __builtin_amdgcn_wmma_f32_16x16x4_f32(false, a, false, bm, (short)0, c,
                                                      false, false);
        }
        int w = col / 3, cch = col % 3;
        #pragma unroll
        for (int jv = 0; jv < 8; ++jv) {
            int i = i0 + jv + (lane >= 16 ? 8 : 0);
            out[base + ((size_t)i * 64 + w) * 6 + 3 + cch] = c[jv];
        }
    }
}

extern "C" void kernel_launch(void* const* d_in, const int* in_sizes, int n_in,
                              void* d_out, int out_size, void* d_ws, size_t ws_size,
                              hipStream_t stream) {
    (void)in_sizes; (void)n_in; (void)out_size; (void)ws_size;
    const float* X  = (const float*)d_in[0];
    const float* w1 = (const float*)d_in[1];
    const float* b1 = (const float*)d_in[2];
    const float* w2 = (const float*)d_in[3];
    const float* b2 = (const float*)d_in[4];
    float* out = (float*)d_out;
    float* ws  = (float*)d_ws;

    float* bp   = ws + WS_BPACK;
    float* vbuf = ws + WS_VBUF;
    float* Mbuf = ws + WS_MBUF;
    float* Kg   = ws + WS_K;

    k_pack_w<<<(NCHUNK * 64 + 255) / 256, 256, 0, stream>>>(w1, w2, bp);
    k_init_v<<<(BATCH * 6 + 255) / 256, 256, 0, stream>>>(b1, b2, vbuf);
    k_init_K<<<16, 256, 0, stream>>>(Kg);

    dim3 g1(BATCH / 16, 4);
    k_proj<<<g1, 256, 0, stream>>>(X, bp, vbuf);
    k_small<<<(BATCH + 255) / 256, 256, 0, stream>>>(vbuf, Mbuf);
    k_rec_hilbert<<<OUTB, 256, 0, stream>>>(X, Mbuf, Kg, out);
}